// TDGNN_75840532512996
// MI455X (gfx1250) — compile-verified
//
#include <hip/hip_runtime.h>
#include <hip/hip_bf16.h>

#define NNODES 100000
#define FIN    512
#define HID    256
#define CLS    64
#define HOPS   3
#define EDGES  3200000

typedef __attribute__((ext_vector_type(2))) float v2f;
typedef __attribute__((ext_vector_type(8))) float v8f;

// ---------------------------------------------------------------------------
// GEMM1: h = relu(x @ W1 + b1)   [100000,512] x [512,256]
// One wave computes a 32(M) x 64(N) tile with V_WMMA_F32_16X16X4_F32.
// Two A fragments (M-blocking) share each B fragment: 8 WMMAs per K-step of 4
// against 2 A b64-loads + 8 B b32-loads (1 B load per WMMA).
// ---------------------------------------------------------------------------
__global__ __launch_bounds__(256) void tdgnn_gemm1(
    const float* __restrict__ x, const float* __restrict__ w,
    const float* __restrict__ bias, float* __restrict__ h) {
  const int wid  = (blockIdx.x * blockDim.x + threadIdx.x) >> 5;
  const int lane = threadIdx.x & 31;
  const int tiles_n = HID / 64;                 // 4
  const int tm = wid / tiles_n;                 // 0..3124  (M tiles of 32)
  const int tn = wid % tiles_n;
  if (tm >= NNODES / 32) return;                // whole-wave exit (EXEC stays full)

  const int m    = lane & 15;                   // A row within 16-row subtile
  const int half = lane >> 4;                   // 0: K0..1, 1: K2..3
  const int kh   = half * 2;
  const int nl   = lane & 15;                   // B/C column within 16
  const int n0   = tn * 64;

  v8f acc0[4], acc1[4];
#pragma unroll
  for (int j = 0; j < 4; ++j) {
    const float bv = bias[n0 + j * 16 + nl];
    acc0[j] = (v8f){bv, bv, bv, bv, bv, bv, bv, bv};
    acc1[j] = acc0[j];
  }

  const float* xrow0 = x + (size_t)(tm * 32 + m) * FIN;        // rows 0..15
  const float* xrow1 = x + (size_t)(tm * 32 + 16 + m) * FIN;   // rows 16..31
  for (int k = 0; k < FIN; k += 4) {
    v2f a0, a1;
    a0.x = xrow0[k + kh];
    a0.y = xrow0[k + kh + 1];
    a1.x = xrow1[k + kh];
    a1.y = xrow1[k + kh + 1];
#pragma unroll
    for (int j = 0; j < 4; ++j) {
      const float* wp = w + (size_t)(k + kh) * HID + n0 + j * 16 + nl;
      v2f b;
      b.x = wp[0];
      b.y = wp[HID];
      acc0[j] = __builtin_amdgcn_wmma_f32_16x16x4_f32(
          false, a0, false, b, (short)0, acc0[j], false, false);
      acc1[j] = __builtin_amdgcn_wmma_f32_16x16x4_f32(
          false, a1, false, b, (short)0, acc1[j], false, false);
    }
  }

  // Epilogue: ReLU + store. C layout: VGPR r -> row r (+8 for lanes 16..31).
#pragma unroll
  for (int j = 0; j < 4; ++j) {
#pragma unroll
    for (int r = 0; r < 8; ++r) {
      const int rr = r + half * 8;
      const float v0 = acc0[j][r];
      const float v1 = acc1[j][r];
      h[(size_t)(tm * 32 + rr) * HID + n0 + j * 16 + nl]      = v0 > 0.f ? v0 : 0.f;
      h[(size_t)(tm * 32 + 16 + rr) * HID + n0 + j * 16 + nl] = v1 > 0.f ? v1 : 0.f;
    }
  }
}

// ---------------------------------------------------------------------------
// GEMM2: z = h @ W2 + b2, and d_out = prop_w[0] * z  (accumulator init)
// One wave computes a 32(M) x 64(N=CLS) tile, same M-blocking as GEMM1.
// ---------------------------------------------------------------------------
__global__ __launch_bounds__(256) void tdgnn_gemm2(
    const float* __restrict__ h, const float* __restrict__ w,
    const float* __restrict__ bias, const float* __restrict__ pw,
    float* __restrict__ z, float* __restrict__ out) {
  const int wid  = (blockIdx.x * blockDim.x + threadIdx.x) >> 5;
  const int lane = threadIdx.x & 31;
  if (wid >= NNODES / 32) return;

  const int m    = lane & 15;
  const int half = lane >> 4;
  const int kh   = half * 2;
  const int nl   = lane & 15;

  v8f acc0[4], acc1[4];
#pragma unroll
  for (int j = 0; j < 4; ++j) {
    const float bv = bias[j * 16 + nl];
    acc0[j] = (v8f){bv, bv, bv, bv, bv, bv, bv, bv};
    acc1[j] = acc0[j];
  }

  const float* hrow0 = h + (size_t)(wid * 32 + m) * HID;
  const float* hrow1 = h + (size_t)(wid * 32 + 16 + m) * HID;
  for (int k = 0; k < HID; k += 4) {
    v2f a0, a1;
    a0.x = hrow0[k + kh];
    a0.y = hrow0[k + kh + 1];
    a1.x = hrow1[k + kh];
    a1.y = hrow1[k + kh + 1];
#pragma unroll
    for (int j = 0; j < 4; ++j) {
      const float* wp = w + (size_t)(k + kh) * CLS + j * 16 + nl;
      v2f b;
      b.x = wp[0];
      b.y = wp[CLS];
      acc0[j] = __builtin_amdgcn_wmma_f32_16x16x4_f32(
          false, a0, false, b, (short)0, acc0[j], false, false);
      acc1[j] = __builtin_amdgcn_wmma_f32_16x16x4_f32(
          false, a1, false, b, (short)0, acc1[j], false, false);
    }
  }

  const float p0 = pw[0];
#pragma unroll
  for (int j = 0; j < 4; ++j) {
#pragma unroll
    for (int r = 0; r < 8; ++r) {
      const int rr = r + half * 8;
      const float v0 = acc0[j][r];
      const float v1 = acc1[j][r];
      const size_t off0 = (size_t)(wid * 32 + rr) * CLS + j * 16 + nl;
      const size_t off1 = (size_t)(wid * 32 + 16 + rr) * CLS + j * 16 + nl;
      z[off0]   = v0;
      out[off0] = p0 * v0;   // full overwrite -> deterministic accumulation base
      z[off1]   = v1;
      out[off1] = p0 * v1;
    }
  }
}

// ---------------------------------------------------------------------------
// Edge propagation: out[dst] += prop_w[l+1] * att[l][e] * z[src]
// 16 lanes per edge, float4 per lane (coalesced 256B gather per edge),
// native f32 global atomics (resolve in L2; z/out both fit in 192MB L2).
// blockIdx.y = hop.
// ---------------------------------------------------------------------------
__global__ __launch_bounds__(256) void tdgnn_edges(
    const int* __restrict__ eidx, const float* __restrict__ att,
    const float* __restrict__ pw, const float* __restrict__ z,
    float* __restrict__ out) {
  const int l = blockIdx.y;
  const int t = blockIdx.x * blockDim.x + threadIdx.x;
  const int e = t >> 4;
  if (e >= EDGES) return;
  const int c4 = (t & 15) * 4;

  const size_t base = (size_t)l * 2 * EDGES;
  const int src = eidx[base + e];
  const int dst = eidx[base + EDGES + e];
  const float wgt = pw[l + 1] * att[(size_t)l * EDGES + e];

  const float4 v = *reinterpret_cast<const float4*>(z + (size_t)src * CLS + c4);
  float* o = out + (size_t)dst * CLS + c4;
  unsafeAtomicAdd(o + 0, wgt * v.x);
  unsafeAtomicAdd(o + 1, wgt * v.y);
  unsafeAtomicAdd(o + 2, wgt * v.z);
  unsafeAtomicAdd(o + 3, wgt * v.w);
}

// ---------------------------------------------------------------------------
// Row-wise log_softmax, in place. One wave32 per row, 2 channels per lane.
// ---------------------------------------------------------------------------
__global__ __launch_bounds__(256) void tdgnn_logsoftmax(float* __restrict__ out) {
  const int row  = (blockIdx.x * blockDim.x + threadIdx.x) >> 5;
  const int lane = threadIdx.x & 31;
  if (row >= NNODES) return;

  float* r = out + (size_t)row * CLS;
  const float a = r[lane];
  const float b = r[lane + 32];

  float mx = fmaxf(a, b);
#pragma unroll
  for (int o = 16; o > 0; o >>= 1) mx = fmaxf(mx, __shfl_xor(mx, o, 32));

  float s = __expf(a - mx) + __expf(b - mx);
#pragma unroll
  for (int o = 16; o > 0; o >>= 1) s += __shfl_xor(s, o, 32);

  const float ls = __logf(s);
  r[lane]      = a - mx - ls;
  r[lane + 32] = b - mx - ls;
}

// ---------------------------------------------------------------------------
extern "C" void kernel_launch(void* const* d_in, const int* in_sizes, int n_in,
                              void* d_out, int out_size, void* d_ws, size_t ws_size,
                              hipStream_t stream) {
  const float* x   = (const float*)d_in[0];
  const float* w1  = (const float*)d_in[1];
  const float* b1  = (const float*)d_in[2];
  const float* w2  = (const float*)d_in[3];
  const float* b2  = (const float*)d_in[4];
  const float* att = (const float*)d_in[5];
  const float* pw  = (const float*)d_in[6];
  const int*  eidx = (const int*)d_in[7];
  float* out = (float*)d_out;

  float* h = (float*)d_ws;                         // [100000, 256]
  float* z = h + (size_t)NNODES * HID;             // [100000, 64]

  // GEMM1: 3125 M-tiles(32) x 4 N-tiles = 12500 waves, 8 waves/block
  tdgnn_gemm1<<<dim3((12500 + 7) / 8), 256, 0, stream>>>(x, w1, b1, h);

  // GEMM2 (+ out = prop_w[0]*z): 3125 wave tiles of 32 rows
  tdgnn_gemm2<<<dim3((3125 + 7) / 8), 256, 0, stream>>>(h, w2, b2, pw, z, out);

  // Edge scatter-add: 3.2M edges * 16 threads / 256 = 200000 blocks, y = hop
  tdgnn_edges<<<dim3(EDGES * 16 / 256, HOPS), 256, 0, stream>>>(eidx, att, pw, z, out);

  // log_softmax: 100000 rows, 8 waves/block
  tdgnn_logsoftmax<<<dim3((NNODES + 7) / 8), 256, 0, stream>>>(out);
}